// SimpleTTTRouter_5059471475438
// MI455X (gfx1250) — compile-verified
//
#include <hip/hip_runtime.h>

typedef float v2f __attribute__((ext_vector_type(2)));
typedef float v8f __attribute__((ext_vector_type(8)));

#define T_TOKENS 16384
#define D_MODEL  4096
#define N_EXP    128
#define TOPK     2

#define BT       64                  // tokens per block
#define KC       32                  // K-chunk staged in LDS
#define KST      36                  // padded LDS row stride in floats (KC + 4)
#define NROWS    (BT + N_EXP)        // 192 staged rows (x tokens then W experts)
#define STAGE_FLOATS (NROWS * KST)   // 6912 floats per stage buffer
#define LOGST    132                 // padded logits row stride
#define SBIAS_OFF (2 * STAGE_FLOATS)          // 13824 floats (beyond both buffers)
#define SMEM_FLOATS (SBIAS_OFF + N_EXP)       // 13952 floats = 55808 B

__global__ __launch_bounds__(256)
void router_gemm_topk(const float* __restrict__ xg,
                      const float* __restrict__ wg,
                      const float* __restrict__ bg,
                      int* __restrict__ idx_out,
                      float* __restrict__ prob_out)
{
    __shared__ float smem[SMEM_FLOATS];
    float* slog  = smem;                 // BT x LOGST, reused after K loop
    float* sbias = smem + SBIAS_OFF;

    const int tid  = threadIdx.x;
    const int wave = tid >> 5;
    const int lane = tid & 31;
    const int half = lane >> 4;          // 0: lanes 0-15, 1: lanes 16-31
    const int l16  = lane & 15;

    const int tokBase = blockIdx.x * BT;

    if (tid < N_EXP) sbias[tid] = bg[tid];

    // wave tiling: 4 token groups (16 rows each) x 2 expert groups (64 cols each)
    const int mw = wave & 3;
    const int nw = wave >> 2;

    // async-copy map: 6 passes x 32 rows; 8 lanes per row cover 32 floats (16B each)
    const int srow0 = tid >> 3;          // 0..31
    const int scol  = (tid & 7) * 4;     // 0..28

    // LDS byte offset of smem[] = low 32 bits of the flat shared-aperture address
    const unsigned smemBase = (unsigned)(unsigned long long)(const void*)smem;

    // Issue 6 per-lane async 16B copies for one K-chunk into stage buffer `buf`.
    auto issue_chunk = [&](int c, int buf) {
        const int kc = c * KC;
        const unsigned bufBase = smemBase + (unsigned)buf * (STAGE_FLOATS * 4u);
        #pragma unroll
        for (int p = 0; p < 6; ++p) {
            const int row = p * 32 + srow0;                 // 0..191
            const float* src = (row < BT)
                ? (xg + (size_t)(tokBase + row) * D_MODEL + kc + scol)
                : (wg + (size_t)(row - BT)     * D_MODEL + kc + scol);
            const unsigned dst = bufBase + (unsigned)(row * KST + scol) * 4u;
            asm volatile("global_load_async_to_lds_b128 %0, %1, off"
                         :: "v"(dst), "v"(src) : "memory");
        }
    };

    v8f acc[4];
    const v8f vzero = {0.f, 0.f, 0.f, 0.f, 0.f, 0.f, 0.f, 0.f};
    #pragma unroll
    for (int n = 0; n < 4; ++n) acc[n] = vzero;

    // WMMA f32 16x16x4 operand offsets (floats) within a stage buffer, per lane:
    // A (16x4): lanes 0-15 hold M=0..15 K={0,1}; lanes 16-31 K={2,3} (contiguous pair)
    const int aoff  = (mw * 16 + l16) * KST + half * 2;
    // B (4x16): lanes 0-15 hold N=0..15 K={0,1}; lanes 16-31 K={2,3}
    const int boff0 = (BT + nw * 64 +  0 + l16) * KST + half * 2;
    const int boff1 = (BT + nw * 64 + 16 + l16) * KST + half * 2;
    const int boff2 = (BT + nw * 64 + 32 + l16) * KST + half * 2;
    const int boff3 = (BT + nw * 64 + 48 + l16) * KST + half * 2;

    issue_chunk(0, 0);

    const int NCHUNK = D_MODEL / KC;     // 128
    for (int c = 0; c < NCHUNK; ++c) {
        // my chunk-c copies have landed in LDS
        asm volatile("s_wait_asynccnt 0x0" ::: "memory");
        // everyone's copies landed; everyone finished computing chunk c-1,
        // so buffer (c+1)&1 is free to refill
        __syncthreads();
        if (c + 1 < NCHUNK) issue_chunk(c + 1, (c + 1) & 1);

        const float* sbuf = smem + (c & 1) * STAGE_FLOATS;
        #pragma unroll
        for (int k0 = 0; k0 < KC; k0 += 4) {
            v2f a  = *(const v2f*)(sbuf + aoff  + k0);
            v2f b0 = *(const v2f*)(sbuf + boff0 + k0);
            acc[0] = __builtin_amdgcn_wmma_f32_16x16x4_f32(false, a, false, b0,
                                                           (short)0, acc[0], false, false);
            v2f b1 = *(const v2f*)(sbuf + boff1 + k0);
            acc[1] = __builtin_amdgcn_wmma_f32_16x16x4_f32(false, a, false, b1,
                                                           (short)0, acc[1], false, false);
            v2f b2 = *(const v2f*)(sbuf + boff2 + k0);
            acc[2] = __builtin_amdgcn_wmma_f32_16x16x4_f32(false, a, false, b2,
                                                           (short)0, acc[2], false, false);
            v2f b3 = *(const v2f*)(sbuf + boff3 + k0);
            acc[3] = __builtin_amdgcn_wmma_f32_16x16x4_f32(false, a, false, b3,
                                                           (short)0, acc[3], false, false);
        }
    }

    // ---- epilogue: spill logits to LDS (reuses stage buffers) ----
    __syncthreads();                     // all waves done reading stage buffers
    const int tokRow = mw * 16 + half * 8;
    #pragma unroll
    for (int n = 0; n < 4; ++n) {
        #pragma unroll
        for (int j = 0; j < 8; ++j) {
            // C/D layout: VGPR j holds M=j (lanes 0-15) / M=j+8 (lanes 16-31), N=l16
            slog[(tokRow + j) * LOGST + nw * 64 + n * 16 + l16] = acc[n][j];
        }
    }
    __syncthreads();

    // ---- softmax + top-2 + renormalize: one thread per token ----
    if (tid < BT) {
        const float* row = slog + tid * LOGST;
        float best   = row[0] + sbias[0];
        float second = -3.402823466e38f;
        int   bi = 0, si = 0;
        for (int e = 1; e < N_EXP; ++e) {
            float v = row[e] + sbias[e];
            if (v > best)        { second = best; si = bi; best = v; bi = e; }
            else if (v > second) { second = v; si = e; }
        }
        float Z = 0.f;
        for (int e = 0; e < N_EXP; ++e) Z += __expf(row[e] + sbias[e] - best);
        float p1 = 1.0f / Z;                     // exp(best-best)/Z
        float p2 = __expf(second - best) / Z;
        float s  = p1 + p2 + 1e-8f;

        const int g = tokBase + tid;
        idx_out[2 * g]      = bi;
        idx_out[2 * g + 1]  = si;
        prob_out[2 * g]     = p1 / s;
        prob_out[2 * g + 1] = p2 / s;
    }
}

extern "C" void kernel_launch(void* const* d_in, const int* in_sizes, int n_in,
                              void* d_out, int out_size, void* d_ws, size_t ws_size,
                              hipStream_t stream) {
    (void)in_sizes; (void)n_in; (void)out_size; (void)d_ws; (void)ws_size;
    const float* x = (const float*)d_in[0];
    const float* W = (const float*)d_in[1];
    const float* b = (const float*)d_in[2];
    // d_out layout: [T*2] int32 top-k indices, then [T*2] f32 top-k probs
    int*   idx_out  = (int*)d_out;
    float* prob_out = (float*)d_out + (size_t)T_TOKENS * TOPK;

    dim3 grid(T_TOKENS / BT);   // 256 blocks
    dim3 block(256);            // 8 wave32 waves
    router_gemm_topk<<<grid, block, 0, stream>>>(x, W, b, idx_out, prob_out);
}